// NLL_76656576299641
// MI455X (gfx1250) — compile-verified
//
#include <hip/hip_runtime.h>
#include <math.h>

#define NN 4096
#define QQ 1024
#define NB 128

typedef float v2f __attribute__((ext_vector_type(2)));
typedef float v8f __attribute__((ext_vector_type(8)));

// ---------------------------------------------------------------------------
// K1: V[i][j] = sig2e*(i==j) + sig2bs0 * exp(-dist[z_i][z_j] / (2*sig2bs1))
// ---------------------------------------------------------------------------
__global__ void build_V(const int* __restrict__ zidx,
                        const float* __restrict__ dist,
                        const float* __restrict__ sig2e,
                        const float* __restrict__ sig2bs,
                        float* __restrict__ V)
{
    long long e = (long long)blockIdx.x * blockDim.x + threadIdx.x;
    int i = (int)(e >> 12);      // / 4096
    int j = (int)(e & 4095);
    float s0 = sig2bs[0];
    float s1 = sig2bs[1];
    float d  = dist[(long long)zidx[i] * QQ + zidx[j]];
    float v  = s0 * expf(-d / (2.0f * s1));
    if (i == j) v += sig2e[0];
    V[e] = v;
}

// ---------------------------------------------------------------------------
// K2: factor the 128x128 diagonal block of panel k in LDS (one workgroup,
//     128 threads, thread t owns row t). Writes back lower triangle + diag.
// ---------------------------------------------------------------------------
__global__ void chol_diag(float* __restrict__ V, int k)
{
    const int P = NB + 1;                 // 129: conflict-avoiding pad
    __shared__ float a[NB * P];
    __shared__ float colL[NB];
    int tid  = threadIdx.x;               // 0..127
    int base = k * NB;

    for (int r = 0; r < NB; ++r)          // coalesced column loads
        a[r * P + tid] = V[(long long)(base + r) * NN + base + tid];
    __syncthreads();

    for (int j = 0; j < NB; ++j) {
        float pivot = sqrtf(a[j * P + j]);
        float lij;
        if (tid == j)      lij = pivot;
        else if (tid > j)  lij = a[tid * P + j] / pivot;
        else               lij = 0.0f;
        colL[tid] = lij;
        __syncthreads();
        if (tid >= j) a[tid * P + j] = lij;
        if (tid > j) {
            for (int c = j + 1; c <= tid; ++c)
                a[tid * P + c] -= lij * colL[c];
        }
        __syncthreads();
    }

    for (int r = 0; r < NB; ++r)
        if (tid <= r)
            V[(long long)(base + r) * NN + base + tid] = a[r * P + tid];
}

// ---------------------------------------------------------------------------
// K3: panel TRSM: L21 = A21 * inv(L11^T). One WG (128 threads) per 128 rows;
//     thread t performs forward substitution for its row, tiles in LDS.
// ---------------------------------------------------------------------------
__global__ void chol_trsm(float* __restrict__ V, int k)
{
    const int P = NB + 1;
    __shared__ float L11[NB * P];
    __shared__ float a[NB * P];
    int tid  = threadIdx.x;               // 0..127
    int base = k * NB;
    long long rowBase = (long long)base + NB + (long long)blockIdx.x * NB;

    for (int r = 0; r < NB; ++r) {
        L11[r * P + tid] = V[(long long)(base + r) * NN + base + tid];
        a[r * P + tid]   = V[(rowBase + r) * NN + base + tid];
    }
    __syncthreads();

    for (int j = 0; j < NB; ++j) {
        float s = a[tid * P + j];
        for (int m = 0; m < j; ++m)
            s -= a[tid * P + m] * L11[j * P + m];   // L11[j][m] is a broadcast
        a[tid * P + j] = s / L11[j * P + j];
    }
    __syncthreads();

    for (int r = 0; r < NB; ++r)
        V[(rowBase + r) * NN + base + tid] = a[r * P + tid];
}

// ---------------------------------------------------------------------------
// K4: trailing update A22 -= L21 * L21^T using V_WMMA_F32_16X16X4_F32.
//     256 threads = 8 waves; wave computes a 32x64 sub-tile of the 128x128
//     output tile = 2x4 grid of 16x16 f32 accumulators. K loop over 128 in
//     steps of 4 (WMMA K=4). Both operand tiles staged in LDS (pad 132:
//     bank = 4*row + col -> conflict-free b64 fragment loads).
// ---------------------------------------------------------------------------
__global__ void chol_syrk(float* __restrict__ V, int k)
{
    const int PK = NB + 4;                // 132
    __shared__ float At[NB * PK];
    __shared__ float Bt[NB * PK];
    int t    = threadIdx.x;               // 0..255
    int base = k * NB;
    long long trail = (long long)base + NB;
    long long R = trail + (long long)blockIdx.x * NB;
    long long S = trail + (long long)blockIdx.y * NB;

    for (int e = t; e < NB * NB; e += 256) {
        int r = e >> 7, c = e & 127;
        At[r * PK + c] = V[(R + r) * NN + base + c];
        Bt[r * PK + c] = V[(S + r) * NN + base + c];
    }
    __syncthreads();

    int wave = t >> 5;                    // 0..7
    int lane = t & 31;
    int lid  = lane & 15;                 // M/N index within fragment
    int half = lane >> 4;                 // 0 or 1
    int k0   = half << 1;                 // K pair start: 0 or 2
    int wm   = wave & 3;                  // 32-row block within tile
    int wn   = wave >> 2;                 // 64-col block within tile

    v8f acc[2][4];
    const v8f vzero = {0.f, 0.f, 0.f, 0.f, 0.f, 0.f, 0.f, 0.f};
#pragma unroll
    for (int ms = 0; ms < 2; ++ms)
#pragma unroll
        for (int ns = 0; ns < 4; ++ns)
            acc[ms][ns] = vzero;

    for (int kk = 0; kk < NB; kk += 4) {
        v2f af[2], bf[4];
#pragma unroll
        for (int ms = 0; ms < 2; ++ms) {
            int row = wm * 32 + ms * 16 + lid;      // A-frag: M varies per lane
            af[ms] = *(const v2f*)&At[row * PK + kk + k0];
        }
#pragma unroll
        for (int ns = 0; ns < 4; ++ns) {
            int row = wn * 64 + ns * 16 + lid;      // B-frag: N varies per lane
            bf[ns] = *(const v2f*)&Bt[row * PK + kk + k0];
        }
#pragma unroll
        for (int ms = 0; ms < 2; ++ms)
#pragma unroll
            for (int ns = 0; ns < 4; ++ns)
                acc[ms][ns] = __builtin_amdgcn_wmma_f32_16x16x4_f32(
                    false, af[ms], false, bf[ns],
                    (short)0, acc[ms][ns], false, false);
    }

    // writeback: A22 -= acc  (C/D layout: VGPR v -> M = v + 8*half, N = lid)
#pragma unroll
    for (int ms = 0; ms < 2; ++ms)
#pragma unroll
        for (int ns = 0; ns < 4; ++ns) {
            long long col = S + wn * 64 + ns * 16 + lid;
#pragma unroll
            for (int v = 0; v < 8; ++v) {
                long long row = R + wm * 32 + ms * 16 + v + 8 * half;
                long long idx = row * NN + col;
                V[idx] = V[idx] - acc[ms][ns][v];
            }
        }
}

// ---------------------------------------------------------------------------
// K5: blocked forward + backward triangular solves (single WG), logdet and
//     quadratic form, final scalar loss.
// ---------------------------------------------------------------------------
__global__ void solve_finalize(const float* __restrict__ yt,
                               const float* __restrict__ yp,
                               const float* __restrict__ V,
                               float* __restrict__ out)
{
    const int P = NB + 1;
    __shared__ float yv[NN];              // solution vector (16 KB)
    __shared__ float Lt[NB * P];          // current diagonal block (66 KB)
    __shared__ float red[256];
    int t = threadIdx.x;                  // 0..255

    for (int i = t; i < NN; i += 256) yv[i] = yt[i] - yp[i];
    __syncthreads();

    // ---- forward: L y = r ----
    for (int b = 0; b < NN / NB; ++b) {
        int base = b * NB;
        for (int e = t; e < NB * NB; e += 256) {
            int r = e >> 7, c = e & 127;
            Lt[r * P + c] = V[(long long)(base + r) * NN + base + c];
        }
        __syncthreads();
        for (int j = 0; j < NB; ++j) {
            float yj = yv[base + j] / Lt[j * P + j];
            __syncthreads();
            if (t == j)                 yv[base + j]  = yj;
            else if (t > j && t < NB)   yv[base + t] -= Lt[t * P + j] * yj;
            __syncthreads();
        }
        for (int i = base + NB + t; i < NN; i += 256) {
            float s = 0.0f;
            const float* row = &V[(long long)i * NN + base];
            for (int m = 0; m < NB; ++m) s += row[m] * yv[base + m];
            yv[i] -= s;
        }
        __syncthreads();
    }

    // ---- backward: L^T x = y (x overwrites yv) ----
    for (int b = NN / NB - 1; b >= 0; --b) {
        int base = b * NB;
        if (t < NB) {                     // gather from already-solved rows
            float s = 0.0f;
            for (long long r = base + NB; r < NN; ++r)
                s += V[r * NN + base + t] * yv[(int)r];
            yv[base + t] -= s;
        }
        for (int e = t; e < NB * NB; e += 256) {
            int r = e >> 7, c = e & 127;
            Lt[r * P + c] = V[(long long)(base + r) * NN + base + c];
        }
        __syncthreads();
        for (int j = NB - 1; j >= 0; --j) {
            float xj = yv[base + j] / Lt[j * P + j];
            __syncthreads();
            if (t == j)       yv[base + j]  = xj;
            else if (t < j)   yv[base + t] -= Lt[j * P + t] * xj;
            __syncthreads();
        }
    }

    // ---- quad = r . x, logdet = 2 * sum(log(L_ii)) ----
    float part = 0.0f, ld = 0.0f;
    for (int i = t; i < NN; i += 256) {
        part += (yt[i] - yp[i]) * yv[i];
        ld   += logf(V[(long long)i * NN + i]);
    }
    red[t] = part;
    __syncthreads();
    for (int s = 128; s > 0; s >>= 1) { if (t < s) red[t] += red[t + s]; __syncthreads(); }
    float quad = red[0];
    __syncthreads();
    red[t] = ld;
    __syncthreads();
    for (int s = 128; s > 0; s >>= 1) { if (t < s) red[t] += red[t + s]; __syncthreads(); }
    if (t == 0) {
        float logdet  = 2.0f * red[0];
        const float log2pi = 1.8378770664093453f;
        out[0] = 0.5f * (float)NN * log2pi + 0.5f * logdet + 0.5f * quad;
    }
}

// ---------------------------------------------------------------------------
extern "C" void kernel_launch(void* const* d_in, const int* in_sizes, int n_in,
                              void* d_out, int out_size, void* d_ws, size_t ws_size,
                              hipStream_t stream)
{
    const float* y_true = (const float*)d_in[0];
    const float* y_pred = (const float*)d_in[1];
    const int*   zidx   = (const int*)d_in[2];
    const float* dist   = (const float*)d_in[3];
    const float* sig2e  = (const float*)d_in[4];
    const float* sig2bs = (const float*)d_in[5];
    float* V = (float*)d_ws;              // 4096*4096*4 = 64 MB scratch

    build_V<<<(NN * (long long)NN) / 256, 256, 0, stream>>>(zidx, dist, sig2e, sig2bs, V);

    for (int k = 0; k < NN / NB; ++k) {
        chol_diag<<<1, NB, 0, stream>>>(V, k);
        int tb = (NN - (k + 1) * NB) / NB;
        if (tb > 0) {
            chol_trsm<<<tb, NB, 0, stream>>>(V, k);
            chol_syrk<<<dim3(tb, tb), 256, 0, stream>>>(V, k);
        }
    }

    solve_finalize<<<1, 256, 0, stream>>>(y_true, y_pred, V, (float*)d_out);
}